// GNN_49752901156905
// MI455X (gfx1250) — compile-verified
//
#include <hip/hip_runtime.h>

typedef __attribute__((ext_vector_type(2))) float v2f;
typedef __attribute__((ext_vector_type(4))) float v4f;
typedef __attribute__((ext_vector_type(8))) float v8f;

#define GCN_N      100000
#define GCN_INDIM  512
#define GCN_H1     128
#define GCN_H2     64
#define GCN_OUT    10

// ---------------------------------------------------------------------------
// Degree / normalization
// ---------------------------------------------------------------------------
__global__ void k_init_deg(float* __restrict__ deg, int n) {
    int i = blockIdx.x * blockDim.x + threadIdx.x;
    if (i < n) deg[i] = 1.0f;                 // self-loop contributes 1
}

__global__ void k_count_deg(const long long* __restrict__ dst,
                            float* __restrict__ deg, int E) {
    int e = blockIdx.x * blockDim.x + threadIdx.x;
    if (e < E) unsafeAtomicAdd(&deg[(int)dst[e]], 1.0f);
}

__global__ void k_rsqrt_inplace(float* __restrict__ deg, int n) {
    int i = blockIdx.x * blockDim.x + threadIdx.x;
    if (i < n) deg[i] = rsqrtf(deg[i]);       // deg >= 1 always
}

// ---------------------------------------------------------------------------
// LDS-staged dense GEMM via V_WMMA_F32_16X16X4_F32: C[M,N] = A[M,K] @ B[K,N]
// Block tile BM x BN (BN == N), WAVES waves, one 16x16 WMMA tile per wave.
// A chunk (BM x BK) is cooperatively staged into LDS transposed (Alds[k][m],
// padded row so WMMA lanes read distinct banks). B stays in L2 (tiny W mats).
// Layouts per CDNA5 ISA 7.12.2:
//   A 16x4:  lanes 0-15 hold M=lane; regs {K=0,1} (low half) / {K=2,3} (high)
//   B 4x16:  lanes hold N=lane&15; reg r -> K = 2*half + r
//   C 16x16: reg r -> M=r (lanes 0-15) / M=r+8 (lanes 16-31), N = lane&15
// ---------------------------------------------------------------------------
template <int BM, int BN, int BK, int WAVES>
__global__ __launch_bounds__(WAVES * 32)
void k_gemm_wmma_lds(const float* __restrict__ A,
                     const float* __restrict__ B,
                     float* __restrict__ C,
                     int M, int N, int K) {
    __shared__ float Alds[BK][BM + 1];        // transposed, padded

    const int tid  = threadIdx.x;
    const int wave = tid >> 5;
    const int lane = tid & 31;
    const int half = lane >> 4;               // 0: K=0..1, 1: K=2..3
    const int l16  = lane & 15;

    const int tilesM = BM >> 4;
    const int wm = wave & (tilesM - 1);       // tile row within block
    const int wn = wave / tilesM;             // tile col within block

    const long long blockRow = (long long)blockIdx.x * BM;
    const int bcol = wn * 16 + l16;
    const int am   = wm * 16 + l16;           // my A row within block tile

    const float* __restrict__ Bp = B + (long long)(half * 2) * N + bcol;

    v8f acc = {};
    for (int k0 = 0; k0 < K; k0 += BK) {
        // ---- cooperative A stage: global row-major -> LDS transposed ----
#pragma unroll
        for (int base = tid * 4; base < BM * BK; base += WAVES * 32 * 4) {
            const int r = base / BK;
            const int c = base % BK;
            v4f v = __builtin_nontemporal_load(
                (const v4f*)(A + (blockRow + r) * K + (k0 + c)));
            Alds[c + 0][r] = v.x;
            Alds[c + 1][r] = v.y;
            Alds[c + 2][r] = v.z;
            Alds[c + 3][r] = v.w;
        }
        __syncthreads();

        // ---- 16x16x4 WMMA over the chunk ----
#pragma unroll
        for (int kk = 0; kk < BK; kk += 4) {
            v2f a, b;
            a.x = Alds[kk + half * 2 + 0][am];
            a.y = Alds[kk + half * 2 + 1][am];
            b.x = Bp[(long long)(k0 + kk) * N];
            b.y = Bp[(long long)(k0 + kk) * N + N];
            acc = __builtin_amdgcn_wmma_f32_16x16x4_f32(
                /*neg_a=*/false, a, /*neg_b=*/false, b,
                /*c_mod=*/(short)0, acc, /*reuse_a=*/false, /*reuse_b=*/false);
        }
        __syncthreads();
    }

    float* __restrict__ Cp =
        C + (blockRow + wm * 16 + half * 8) * N + bcol;
#pragma unroll
    for (int r = 0; r < 8; ++r)
        Cp[(long long)r * N] = acc[r];
}

// ---------------------------------------------------------------------------
// agg[i][c] = bias[c] + h[i][c] * dinv[i]^2     (self-loop + bias)
// ---------------------------------------------------------------------------
__global__ void k_aggr_init(const float* __restrict__ h,
                            const float* __restrict__ dinv,
                            const float* __restrict__ bias,
                            float* __restrict__ agg,
                            int total, int shift) {
    int t = blockIdx.x * blockDim.x + threadIdx.x;
    if (t >= total) return;
    const int i = t >> shift;
    const int c = t & ((1 << shift) - 1);
    const float di = dinv[i];
    agg[t] = bias[c] + h[t] * (di * di);
}

// ---------------------------------------------------------------------------
// Edge scatter: wave-per-edge grid-stride loop, CPT channels/lane (32*CPT==C).
// agg[dst][c] += h[src][c] * dinv[src]*dinv[dst]
// One-iteration-ahead prefetch of the next gathered row hides the dependent
// chain (src[e] -> h[src] row) behind the current edge's atomics.
// ---------------------------------------------------------------------------
template <int C, int CPT>
__global__ void k_scatter(const long long* __restrict__ src,
                          const long long* __restrict__ dst,
                          const float* __restrict__ dinv,
                          const float* __restrict__ h,
                          float* __restrict__ agg, int E) {
    const int wavesPerBlock = blockDim.x >> 5;
    const int wavesTotal = gridDim.x * wavesPerBlock;
    const int w0 = blockIdx.x * wavesPerBlock + (threadIdx.x >> 5);
    const int lane = threadIdx.x & 31;
    const int c0 = lane * CPT;

    for (int e = w0; e < E; e += wavesTotal) {      // uniform per wave
        const int en = e + wavesTotal;
        if (en < E) {
            const int sn = (int)src[en];
            __builtin_prefetch(h + (long long)sn * C + c0, 0, 0);
        }
        const int s = (int)src[e];
        const int d = (int)dst[e];
        const float norm = dinv[s] * dinv[d];

        const float* __restrict__ hs = h   + (long long)s * C + c0;
        float* __restrict__       ad = agg + (long long)d * C + c0;
#pragma unroll
        for (int j = 0; j < CPT; ++j)
            unsafeAtomicAdd(ad + j, hs[j] * norm);
    }
}

__global__ void k_relu(float* __restrict__ x, int total) {
    int t = blockIdx.x * blockDim.x + threadIdx.x;
    if (t < total) x[t] = fmaxf(x[t], 0.0f);
}

// ---------------------------------------------------------------------------
// Head: out[i][o] = sum_k h[i][k] * Wl[k][o] + bl[o]   (K=64, O=10)
// ---------------------------------------------------------------------------
__global__ void k_head(const float* __restrict__ h,
                       const float* __restrict__ Wl,
                       const float* __restrict__ bl,
                       float* __restrict__ out, int n) {
    int i = blockIdx.x * blockDim.x + threadIdx.x;
    if (i >= n) return;
    float acc[GCN_OUT];
#pragma unroll
    for (int o = 0; o < GCN_OUT; ++o) acc[o] = bl[o];
    const float* __restrict__ hr = h + (long long)i * GCN_H2;
#pragma unroll 4
    for (int k = 0; k < GCN_H2; ++k) {
        const float hv = hr[k];
#pragma unroll
        for (int o = 0; o < GCN_OUT; ++o)
            acc[o] += hv * Wl[k * GCN_OUT + o];
    }
    float* __restrict__ op = out + (long long)i * GCN_OUT;
#pragma unroll
    for (int o = 0; o < GCN_OUT; ++o) op[o] = acc[o];
}

// ---------------------------------------------------------------------------
extern "C" void kernel_launch(void* const* d_in, const int* in_sizes, int n_in,
                              void* d_out, int out_size, void* d_ws, size_t ws_size,
                              hipStream_t stream) {
    const float*     x   = (const float*)d_in[0];
    const long long* ei  = (const long long*)d_in[1];   // int64 [2, E]
    const float*     W1  = (const float*)d_in[2];
    const float*     b1  = (const float*)d_in[3];
    const float*     W2  = (const float*)d_in[4];
    const float*     b2  = (const float*)d_in[5];
    const float*     Wl  = (const float*)d_in[6];
    const float*     bl  = (const float*)d_in[7];
    float*           out = (float*)d_out;

    const int N = GCN_N;
    const int E = in_sizes[1] / 2;
    const long long* esrc = ei;
    const long long* edst = ei + E;

    // Workspace layout (256B aligned)
    char* ws = (char*)d_ws;
    size_t off = 0;
    auto alloc = [&](size_t bytes) {
        char* p = ws + off;
        off += (bytes + 255) & ~(size_t)255;
        return p;
    };
    float* dinv = (float*)alloc((size_t)N * 4);                 // deg -> dinv
    float* bufA = (float*)alloc((size_t)N * GCN_H1 * 4);        // gemm outputs
    float* bufB = (float*)alloc((size_t)N * GCN_H1 * 4);        // aggregates
    (void)ws_size;

    const int TPB = 256;
    const int SCATTER_BLOCKS = 4096;                            // grid-stride

    // ---- normalization ----
    k_init_deg<<<(N + TPB - 1) / TPB, TPB, 0, stream>>>(dinv, N);
    k_count_deg<<<(E + TPB - 1) / TPB, TPB, 0, stream>>>(edst, dinv, E);
    k_rsqrt_inplace<<<(N + TPB - 1) / TPB, TPB, 0, stream>>>(dinv, N);

    // ---- layer 1: h = x @ W1 ; agg = b1 + selfloop + scatter ; relu ----
    {
        // 16x128 block tile, 8 waves (1x8 tiles), K-chunk 64
        k_gemm_wmma_lds<16, GCN_H1, 64, 8>
            <<<N / 16, 256, 0, stream>>>(x, W1, bufA, N, GCN_H1, GCN_INDIM);

        const int total = N * GCN_H1;
        k_aggr_init<<<(total + TPB - 1) / TPB, TPB, 0, stream>>>(bufA, dinv, b1, bufB, total, 7);

        k_scatter<GCN_H1, 4><<<SCATTER_BLOCKS, TPB, 0, stream>>>(esrc, edst, dinv, bufA, bufB, E);

        k_relu<<<(total + TPB - 1) / TPB, TPB, 0, stream>>>(bufB, total);
    }

    // ---- layer 2: h = h1 @ W2 ; aggregate ; relu ----
    {
        // 32x64 block tile, 8 waves (2x4 tiles), K-chunk 64
        k_gemm_wmma_lds<32, GCN_H2, 64, 8>
            <<<N / 32, 256, 0, stream>>>(bufB, W2, bufA, N, GCN_H2, GCN_H1);

        const int total = N * GCN_H2;
        k_aggr_init<<<(total + TPB - 1) / TPB, TPB, 0, stream>>>(bufA, dinv, b2, bufB, total, 6);

        k_scatter<GCN_H2, 2><<<SCATTER_BLOCKS, TPB, 0, stream>>>(esrc, edst, dinv, bufA, bufB, E);

        k_relu<<<(total + TPB - 1) / TPB, TPB, 0, stream>>>(bufB, total);
    }

    // ---- head ----
    k_head<<<(N + TPB - 1) / TPB, TPB, 0, stream>>>(bufB, Wl, bl, out, N);
    (void)out_size; (void)n_in;
}